// YoloLoss_30030411333646
// MI455X (gfx1250) — compile-verified
//
#include <hip/hip_runtime.h>
#include <hip/hip_bf16.h>

// Problem constants (from reference): B=16, nA=3, R=C=64, nC=80, nT=512.
// Output layout (floats, concatenated in return order):
//   [0)        noobj_mask  16*3*64*64   = 196608    (init 1.0)
//   [196608)   obj_mask    16*3*64*64   = 196608    (init 0.0)
//   [393216)   tcls        16*3*64*64*80= 15728640  (init 0.0)
// Total 16121856 floats (~64.5 MB) => pure store-bandwidth bound (~2.8us @ 23.3 TB/s).

#define NOOBJ_ELEMS   196608
#define OBJ_OFFSET    196608
#define TCLS_OFFSET   393216

// ---------------------------------------------------------------------------
// Bulk fill via CDNA5 async LDS->global DMA stores (ASYNCcnt-tracked).
// LDS: bytes [0,512) = 1.0f pattern, [512,1024) = 0.0f pattern.
// Each lane issues 4 x global_store_async_from_lds_b128 per iteration
// (64 B per lane per iteration via the immediate offset field); data flows
// LDS -> L2 -> HBM with no VGPR data registers and up to 63 stores in flight.
// n64 = number of 64-byte groups per thread-stride walk; pattern_off selects
// which 512B LDS pattern feeds the stores.
// ---------------------------------------------------------------------------
__global__ void yolo_fill_async(float* __restrict__ base,
                                unsigned n64,          // count of 64B groups
                                unsigned pattern_off)  // 0 = ones, 512 = zeros
{
    __shared__ float buf[256];          // single shared array => LDS offset 0
    int tid = threadIdx.x;
    buf[tid] = (tid < 128) ? 1.0f : 0.0f;
    __syncthreads();

    unsigned lds = (((unsigned)tid & 31u) << 4) + pattern_off;

    unsigned i      = blockIdx.x * blockDim.x + tid;
    unsigned stride = gridDim.x * blockDim.x;
    unsigned long long b = (unsigned long long)base;

    for (; i < n64; i += stride) {
        unsigned long long addr = b + ((unsigned long long)(i << 6));
        asm volatile(
            "global_store_async_from_lds_b128 %0, %1, off\n\t"
            "global_store_async_from_lds_b128 %0, %1, off offset:16\n\t"
            "global_store_async_from_lds_b128 %0, %1, off offset:32\n\t"
            "global_store_async_from_lds_b128 %0, %1, off offset:48"
            :: "v"(addr), "v"(lds) : "memory");
    }
    // Drain the async DMA queue before the wave retires.
    asm volatile("s_wait_asynccnt 0x0" ::: "memory");
}

// ---------------------------------------------------------------------------
// Per-target scatter (512 targets). All writes are idempotent constants,
// so plain stores suffice (no atomics).
// ---------------------------------------------------------------------------
__global__ void yolo_scatter(const float* __restrict__ target,
                             float* __restrict__ out,
                             int nT) {
    int t = blockIdx.x * blockDim.x + threadIdx.x;
    if (t >= nT) return;

    const float S = 64.0f;  // grid scale (R == C == 64)
    float cls_f = target[t * 6 + 1];
    float gx    = target[t * 6 + 2] * S;
    float gy    = target[t * 6 + 3] * S;
    float gw    = target[t * 6 + 4] * S;
    float gh    = target[t * 6 + 5] * S;

    int gi  = (int)gx;          // truncation == floor (positive range)
    int gj  = (int)gy;
    int cls = (int)cls_f;
    int b   = t & 15;           // image_id = t % 16 (jnp.tile(arange(16), 32))

    // Scaled anchors: [0.05,0.07,0.12,0.18,0.3,0.35] reshaped (3,2) * 64
    const float aw[3] = { 3.2f,  7.68f, 19.2f };
    const float ah[3] = { 4.48f, 11.52f, 22.4f };

    float iou[3];
    #pragma unroll
    for (int a = 0; a < 3; ++a) {
        float inter = fminf(aw[a], gw) * fminf(ah[a], gh);
        float uni   = aw[a] * ah[a] + 1e-16f + gw * gh - inter;
        iou[a] = inter / uni;
    }
    // argmax, first-occurrence tie-break (matches jnp.argmax)
    int best = 0;
    if (iou[1] > iou[best]) best = 1;
    if (iou[2] > iou[best]) best = 2;

    float* noobj = out;
    float* obj   = out + OBJ_OFFSET;
    float* tcls  = out + TCLS_OFFSET;

    // flat index into (16, 3, 64, 64)
    int ib = ((b * 3 + best) * 64 + gi) * 64 + gj;
    obj[ib]   = 1.0f;
    noobj[ib] = 0.0f;

    #pragma unroll
    for (int a = 0; a < 3; ++a) {
        if (iou[a] > 0.5f) {
            int ia = ((b * 3 + a) * 64 + gi) * 64 + gj;
            noobj[ia] = 0.0f;
        }
    }

    tcls[(long long)ib * 80 + cls] = 1.0f;
}

extern "C" void kernel_launch(void* const* d_in, const int* in_sizes, int n_in,
                              void* d_out, int out_size, void* d_ws, size_t ws_size,
                              hipStream_t stream) {
    (void)in_sizes; (void)n_in; (void)d_ws; (void)ws_size; (void)out_size;
    // d_in[0] = x (shape-only in the reference; data unused)
    const float* target = (const float*)d_in[1];
    float* out = (float*)d_out;

    // Region 1: noobj = 1.0f  -> 196608 floats = 12288 x 64B groups
    // Region 2: obj+tcls = 0  -> 15925248 floats = 995328 x 64B groups
    yolo_fill_async<<<48,   256, 0, stream>>>(out,              12288u,  0u);
    yolo_fill_async<<<3888, 256, 0, stream>>>(out + OBJ_OFFSET, 995328u, 512u);

    // Scatter: 512 targets, one lane each.
    yolo_scatter<<<2, 256, 0, stream>>>(target, out, 512);
}